// SparseFingerprint_TS_DRSN_29506425323874
// MI455X (gfx1250) — compile-verified
//
#include <hip/hip_runtime.h>
#include <math.h>

typedef float f4  __attribute__((ext_vector_type(4)));
typedef float v2f __attribute__((ext_vector_type(2)));
typedef float v8f __attribute__((ext_vector_type(8)));

static constexpr int Bn = 8192;
static constexpr int Cc = 100;
static constexpr int Kk = 8;
static constexpr int Dd = 4096;

// ---------------------------------------------------------------------------
// Kernel A: logits = raw / (softplus(log_temp)+1e-3); per-row argmax + maxprob.
// One wave (32 lanes) per row; blockDim = 256 -> 8 rows per block.
// ---------------------------------------------------------------------------
__global__ void sf_logits_kernel(const float* __restrict__ raw,
                                 const float* __restrict__ log_temp,
                                 float* __restrict__ logits_out,
                                 int*   __restrict__ pred,
                                 float* __restrict__ feat1)
{
    const int lane = threadIdx.x & 31;
    const int wave = threadIdx.x >> 5;
    const int row  = blockIdx.x * 8 + wave;
    if (row >= Bn) return;

    // stable softplus
    const float lt    = log_temp[0];
    const float temp  = fmaxf(lt, 0.0f) + log1pf(expf(-fabsf(lt))) + 0.001f;
    const float rtemp = 1.0f / temp;

    const float* rrow = raw + (size_t)row * Cc;
    float*       orow = logits_out + (size_t)row * Cc;

    float lv[4];
    int   cs[4];
    int   n  = 0;
    float m  = -3.4e38f;
    int   mi = 0x7fffffff;
    for (int c = lane; c < Cc; c += 32) {
        float l = rrow[c] * rtemp;
        orow[c] = l;
        lv[n] = l; cs[n] = c; ++n;
        if (l > m) { m = l; mi = c; }
    }
    // wave32 argmax reduce (first-index tie-break, matches jnp.argmax)
    for (int off = 16; off; off >>= 1) {
        float om = __shfl_xor(m,  off, 32);
        int   oi = __shfl_xor(mi, off, 32);
        if (om > m || (om == m && oi < mi)) { m = om; mi = oi; }
    }
    // max softmax prob = 1 / sum(exp(l - max))
    float se = 0.0f;
    for (int i = 0; i < n; ++i) se += expf(lv[i] - m);
    for (int off = 16; off; off >>= 1) se += __shfl_xor(se, off, 32);

    if (lane == 0) {
        pred[row]  = mi;
        feat1[row] = 1.0f - 1.0f / se;   // 1 - max_prob
    }
}

// ---------------------------------------------------------------------------
// Kernel B (HBM-bound): ham[b] = min_k mean_d |code[b,d] - cent[pred[b],k,d]|
// One 256-thread block per row. code row streamed once with non-temporal
// B128 loads (don't pollute L2 -> centroids stay L2-resident); each loaded
// code float4 is reused 8x against the K centroid rows from registers.
// ---------------------------------------------------------------------------
__global__ void sf_ham_kernel(const float* __restrict__ code,
                              const float* __restrict__ cents,
                              const int*   __restrict__ pred,
                              float*       __restrict__ ham)
{
    const int b    = blockIdx.x;
    const int tid  = threadIdx.x;
    const int lane = tid & 31;
    const int wave = tid >> 5;

    const float* crow  = code  + (size_t)b * Dd;
    const float* cbase = cents + (size_t)pred[b] * (Kk * Dd);

    float acc[Kk] = {};
    for (int d = tid * 4; d < Dd; d += 256 * 4) {
        if (d + 1024 < Dd) __builtin_prefetch(crow + d + 1024, 0, 0);
        f4 cv = __builtin_nontemporal_load((const f4*)(crow + d));
#pragma unroll
        for (int k = 0; k < Kk; ++k) {
            f4 ev = *(const f4*)(cbase + k * Dd + d);
            acc[k] += fabsf(cv.x - ev.x) + fabsf(cv.y - ev.y) +
                      fabsf(cv.z - ev.z) + fabsf(cv.w - ev.w);
        }
    }

    __shared__ float red[8][Kk];
#pragma unroll
    for (int k = 0; k < Kk; ++k) {
        float v = acc[k];
        for (int off = 16; off; off >>= 1) v += __shfl_xor(v, off, 32);
        if (lane == 0) red[wave][k] = v;
    }
    __syncthreads();

    if (tid == 0) {
        float best = 3.4e38f;
#pragma unroll
        for (int k = 0; k < Kk; ++k) {
            float s = 0.0f;
            for (int w = 0; w < 8; ++w) s += red[w][k];
            best = fminf(best, s * (1.0f / Dd));
        }
        ham[b] = best;
    }
}

// ---------------------------------------------------------------------------
// Kernel C: calibration MLP via V_WMMA_F32_16X16X4_F32.
//   A (16x4):  M=rows (16 per wave), K0=ham, K1=1-maxprob, K2/K3 = 0 pad
//              lanes 0-15 hold K0/K1 in VGPR0/1; lanes 16-31 hold K2/K3 (=0)
//   B (4x16):  N=hidden unit (8 real + 8 zero pad), same K striping
//   C/D(16x16): VGPR r = M=r (lanes 0-15) / M=r+8 (lanes 16-31), N = lane&15
// Then relu, second layer as a 16-lane butterfly reduce over N, sigmoid.
// blockDim=256 (8 waves) -> 128 rows per block.
// ---------------------------------------------------------------------------
__global__ void sf_mlp_wmma_kernel(const float* __restrict__ ham,
                                   const float* __restrict__ feat1,
                                   const float* __restrict__ w1,   // [8][2]
                                   const float* __restrict__ b1,   // [8]
                                   const float* __restrict__ w2,   // [8]
                                   const float* __restrict__ b2,   // [1]
                                   float* __restrict__ out)        // [B]
{
    const int lane = threadIdx.x & 31;
    const int wave = threadIdx.x >> 5;
    const int row0 = (blockIdx.x * 8 + wave) * 16;
    const int lm   = lane & 15;
    const bool hi  = lane >= 16;

    v2f a;  a.x = 0.0f;  a.y = 0.0f;
    if (!hi) { a.x = ham[row0 + lm]; a.y = feat1[row0 + lm]; }  // K=0, K=1

    v2f bm; bm.x = 0.0f; bm.y = 0.0f;
    if (!hi && lm < 8) { bm.x = w1[lm * 2 + 0]; bm.y = w1[lm * 2 + 1]; }

    const float bias = (lm < 8) ? b1[lm] : 0.0f;
    v8f c;
#pragma unroll
    for (int r = 0; r < 8; ++r) c[r] = bias;

    // D = A x B + C  (16x16 f32 accumulator, 8 VGPRs)
    v8f d = __builtin_amdgcn_wmma_f32_16x16x4_f32(
        /*neg_a=*/false, a, /*neg_b=*/false, bm,
        /*c_mod=*/(short)0, c, /*reuse_a=*/false, /*reuse_b=*/false);

    const float w2n = (lm < 8) ? w2[lm] : 0.0f;
    const float b2v = b2[0];
#pragma unroll
    for (int r = 0; r < 8; ++r) {
        float t = fmaxf(d[r], 0.0f) * w2n;      // relu * second-layer weight
        t += __shfl_xor(t, 8, 16);              // reduce over N within 16-lane half
        t += __shfl_xor(t, 4, 16);
        t += __shfl_xor(t, 2, 16);
        t += __shfl_xor(t, 1, 16);
        if (lm == 0) {
            const int row = row0 + r + (hi ? 8 : 0);
            out[row] = 1.0f / (1.0f + expf(-(t + b2v)));
        }
    }
}

// ---------------------------------------------------------------------------
extern "C" void kernel_launch(void* const* d_in, const int* in_sizes, int n_in,
                              void* d_out, int out_size, void* d_ws, size_t ws_size,
                              hipStream_t stream)
{
    const float* code = (const float*)d_in[0];   // [B, D]
    const float* raw  = (const float*)d_in[1];   // [B, C]
    const float* cent = (const float*)d_in[2];   // [C, K, D]
    const float* w1   = (const float*)d_in[3];   // [8, 2]
    const float* b1   = (const float*)d_in[4];   // [8]
    const float* w2   = (const float*)d_in[5];   // [1, 8]
    const float* b2   = (const float*)d_in[6];   // [1]
    const float* lt   = (const float*)d_in[7];   // [1]

    float* logits_out = (float*)d_out;                       // [B, C]
    float* score_out  = logits_out + (size_t)Bn * Cc;        // [B]

    int*   pred  = (int*)d_ws;                               // [B]
    float* feat1 = (float*)((int*)d_ws + Bn);                // [B]  (1 - max_prob)
    float* hamv  = feat1 + Bn;                               // [B]

    sf_logits_kernel<<<Bn / 8, 256, 0, stream>>>(raw, lt, logits_out, pred, feat1);
    sf_ham_kernel<<<Bn, 256, 0, stream>>>(code, cent, pred, hamv);
    sf_mlp_wmma_kernel<<<Bn / 128, 256, 0, stream>>>(hamv, feat1, w1, b1, w2, b2, score_out);
}